// bidirLSTM_78709570667078
// MI455X (gfx1250) — compile-verified
//
#include <hip/hip_runtime.h>
#include <hip/hip_bf16.h>

// ---------------------------------------------------------------------------
// Bidirectional LSTM, T=8192, E=D=60, gates G=4*D=240, classes C=5.
//
// K1: gx[t][g] = sum_e emb[tokens[t]][e] * Wxh[g][e] + (bxh[g]+bhh[g])
//     done as a WMMA f32 16x16x4 GEMM (M=8192, K=60 -> 15 k-steps, N=240).
// K2: sequential scan per direction (1 workgroup each), Whh in registers,
//     h broadcast through LDS.
// K3: out[c] = Why_L[c,:] @ hL + Why_R[c,:] @ hR
// ---------------------------------------------------------------------------

typedef float v2f __attribute__((ext_vector_type(2)));
typedef float v8f __attribute__((ext_vector_type(8)));

#define T_LEN 8192
#define E_DIM 60
#define D_DIM 60
#define G_DIM 240   // 4*D
#define C_DIM 5

// ---------------------------------------------------------------------------
// Kernel 1: gather + GEMM via V_WMMA_F32_16X16X4_F32.
// Wave task: (dir, ntile 0..14, mchunk 0..31); each mchunk = 16 M-tiles (256 rows).
// 2*15*32 = 960 waves -> 120 blocks of 256 threads (8 waves).
//
// ISA 7.12.2 layouts (f32):
//  A 16x4 : lanes 0-15 -> row M=lane, V0=K0,V1=K1 ; lanes 16-31 -> V0=K2,V1=K3
//  B 4x16 : lanes 0-15 -> col N=lane, V0=K0,V1=K1 ; lanes 16-31 -> V0=K2,V1=K3
//  C 16x16: VGPR r -> (M=r, N=lane) for lanes 0-15 ; (M=r+8, N=lane-16) else
// So per k-step each lane loads one float2 from row-major storage at column
// k0 + (lane>=16 ? 2 : 0) -- identical addressing for A (x rows) and B (Wxh rows).
// ---------------------------------------------------------------------------
__global__ __launch_bounds__(256)
void bilstm_gatesx_wmma(const int* __restrict__ tokens,
                        const float* __restrict__ emb,
                        const float* __restrict__ WxhL,
                        const float* __restrict__ bxhL,
                        const float* __restrict__ bhhL,
                        const float* __restrict__ WxhR,
                        const float* __restrict__ bxhR,
                        const float* __restrict__ bhhR,
                        float* __restrict__ gxL,
                        float* __restrict__ gxR) {
    const int wave = (blockIdx.x * blockDim.x + threadIdx.x) >> 5;
    const int lane = threadIdx.x & 31;
    const int dir  = wave / (15 * 32);
    const int rem  = wave % (15 * 32);
    const int nt   = rem / 32;   // N tile (16 gates)
    const int mc   = rem % 32;   // chunk of 16 M-tiles

    const float* __restrict__ Wxh = dir ? WxhR : WxhL;
    const float* __restrict__ bx  = dir ? bxhR : bxhL;
    const float* __restrict__ bh  = dir ? bhhR : bhhL;
    float* __restrict__ gx        = dir ? gxR  : gxL;

    const int n0   = nt * 16;
    const int l15  = lane & 15;
    const int khi  = (lane >> 4) * 2;       // 0 for lanes 0-15, 2 for 16-31
    const int rowB = (lane >> 4) * 8;       // C-tile row offset per half-wave

    // B = Wxh.T slice for this N tile: 15 float2 per lane, reused for 16 M tiles.
    v2f Bm[15];
    const float* wrow = Wxh + (n0 + l15) * E_DIM;
    #pragma unroll
    for (int k = 0; k < 15; ++k)
        Bm[k] = *reinterpret_cast<const v2f*>(wrow + 4 * k + khi);

    const float bias = bx[n0 + l15] + bh[n0 + l15];   // per-column bias

    for (int mt = 0; mt < 16; ++mt) {
        const int m0  = (mc * 16 + mt) * 16;
        const int row = m0 + l15;
        const long tok = (long)tokens[row];
        const float* xrow = emb + tok * E_DIM;

        v8f acc = {};
        #pragma unroll
        for (int k = 0; k < 15; ++k) {
            v2f a = *reinterpret_cast<const v2f*>(xrow + 4 * k + khi);
            acc = __builtin_amdgcn_wmma_f32_16x16x4_f32(
                      false, a, false, Bm[k], (short)0, acc, false, false);
        }
        #pragma unroll
        for (int r = 0; r < 8; ++r)
            gx[(long)(m0 + rowB + r) * G_DIM + n0 + l15] = acc[r] + bias;
    }
}

// ---------------------------------------------------------------------------
// Kernel 2: sequential LSTM scan. grid = 2 blocks (dir), 256 threads.
// Thread g<240 owns Whh row g in 60 VGPRs. h lives in LDS (broadcast reads).
// Gate order per reference jnp.split: i, f, cg, o.
// ---------------------------------------------------------------------------
__device__ __forceinline__ float fast_sigmoid(float x) {
    return 1.0f / (1.0f + __expf(-x));
}
__device__ __forceinline__ float fast_tanh(float x) {
    // tanh(x) = 1 - 2/(exp(2x)+1)
    return 1.0f - 2.0f / (__expf(2.0f * x) + 1.0f);
}

__global__ __launch_bounds__(256)
void bilstm_scan(const float* __restrict__ gxL,
                 const float* __restrict__ gxR,
                 const float* __restrict__ WhhL,
                 const float* __restrict__ WhhR,
                 float* __restrict__ hOutL,
                 float* __restrict__ hOutR) {
    __shared__ __align__(16) float h_sm[64];      // 60 used, padded for float4
    __shared__ float gates_sm[G_DIM];

    const int dir = blockIdx.x;
    const int tid = threadIdx.x;
    const float* __restrict__ Whh = dir ? WhhR : WhhL;
    const float* __restrict__ gx  = dir ? gxR  : gxL;

    // Pin this gate's Whh row in registers.
    float w[D_DIM];
    if (tid < G_DIM) {
        #pragma unroll
        for (int e = 0; e < D_DIM; ++e) w[e] = Whh[tid * D_DIM + e];
    }
    float cell = 0.0f;
    if (tid < 64) h_sm[tid] = 0.0f;
    __syncthreads();

    for (int t = 0; t < T_LEN; ++t) {
        const int ti = dir ? (T_LEN - 1 - t) : t;   // right LSTM consumes reversed
        if (tid < G_DIM) {
            float acc = gx[(long)ti * G_DIM + tid];
            if (t + 4 < T_LEN) {
                const int tn = dir ? (T_LEN - 5 - t) : (t + 4);
                __builtin_prefetch(gx + (long)tn * G_DIM + tid, 0, 0);
            }
            const float4* h4 = reinterpret_cast<const float4*>(h_sm);
            #pragma unroll
            for (int e4 = 0; e4 < 15; ++e4) {
                float4 hv = h4[e4];                 // ds_load_b128, broadcast
                acc += w[4 * e4 + 0] * hv.x;
                acc += w[4 * e4 + 1] * hv.y;
                acc += w[4 * e4 + 2] * hv.z;
                acc += w[4 * e4 + 3] * hv.w;
            }
            gates_sm[tid] = acc;
        }
        __syncthreads();
        if (tid < D_DIM) {
            const float ig = fast_sigmoid(gates_sm[tid]);
            const float fg = fast_sigmoid(gates_sm[D_DIM + tid]);
            const float cg = fast_tanh(gates_sm[2 * D_DIM + tid]);
            const float og = fast_sigmoid(gates_sm[3 * D_DIM + tid]);
            cell = fg * cell + ig * cg;
            h_sm[tid] = og * fast_tanh(cell);
        }
        __syncthreads();
    }

    if (tid < D_DIM) (dir ? hOutR : hOutL)[tid] = h_sm[tid];
}

// ---------------------------------------------------------------------------
// Kernel 3: out[c] = Why_L[c,:] @ hL + Why_R[c,:] @ hR
// ---------------------------------------------------------------------------
__global__ void bilstm_combine(const float* __restrict__ WhyL,
                               const float* __restrict__ WhyR,
                               const float* __restrict__ hL,
                               const float* __restrict__ hR,
                               float* __restrict__ out) {
    const int c = threadIdx.x;
    if (c < C_DIM) {
        float s = 0.0f;
        #pragma unroll
        for (int d = 0; d < D_DIM; ++d)
            s += WhyL[c * D_DIM + d] * hL[d] + WhyR[c * D_DIM + d] * hR[d];
        out[c] = s;
    }
}

// ---------------------------------------------------------------------------
extern "C" void kernel_launch(void* const* d_in, const int* in_sizes, int n_in,
                              void* d_out, int out_size, void* d_ws, size_t ws_size,
                              hipStream_t stream) {
    const int*   tokens = (const int*)  d_in[0];
    const float* emb    = (const float*)d_in[1];
    const float* WxhL   = (const float*)d_in[2];
    const float* WhhL   = (const float*)d_in[3];
    const float* bxhL   = (const float*)d_in[4];
    const float* bhhL   = (const float*)d_in[5];
    const float* WxhR   = (const float*)d_in[6];
    const float* WhhR   = (const float*)d_in[7];
    const float* bxhR   = (const float*)d_in[8];
    const float* bhhR   = (const float*)d_in[9];
    const float* WhyL   = (const float*)d_in[10];
    const float* WhyR   = (const float*)d_in[11];
    float* out = (float*)d_out;

    // Workspace layout: gxL | gxR | hL | hR
    float* gxL = (float*)d_ws;
    float* gxR = gxL + (size_t)T_LEN * G_DIM;
    float* hL  = gxR + (size_t)T_LEN * G_DIM;
    float* hR  = hL + 64;

    // K1: 960 waves -> 120 blocks * 256 threads
    bilstm_gatesx_wmma<<<120, 256, 0, stream>>>(tokens, emb,
                                                WxhL, bxhL, bhhL,
                                                WxhR, bxhR, bhhR,
                                                gxL, gxR);
    // K2: one workgroup per direction
    bilstm_scan<<<2, 256, 0, stream>>>(gxL, gxR, WhhL, WhhR, hL, hR);
    // K3: tiny combine
    bilstm_combine<<<1, 32, 0, stream>>>(WhyL, WhyR, hL, hR, out);
}